// CoExpert_49168785604634
// MI455X (gfx1250) — compile-verified
//
#include <hip/hip_runtime.h>
#include <hip/hip_bf16.h>
#include <math.h>

// ---------------------------------------------------------------------------
// Problem: C=64 cliques, S=16, D=256, P=65536
//   W[c] = softmax( cos-sim(reps[c]) )            [64,16,16]
//   out[c] = W[c] @ params[c]                      [64,16,65536]
// Memory-bound: ~537 MB traffic -> ~23us floor @ 23.3 TB/s.
// GEMM done with V_WMMA_F32_16X16X4_F32 (f32-exact, matches reference dtype).
// ---------------------------------------------------------------------------

typedef __attribute__((ext_vector_type(2))) float v2f;
typedef __attribute__((ext_vector_type(8))) float v8f;

#define C_CLIQUES 64
#define S_SIZE    16
#define D_DIM     256
#define P_DIM     65536
#define EPSF      1e-8f

// ---------------------------------------------------------------------------
// Kernel 1: per-clique cosine-similarity attention weights (tiny: 64 x 16x16)
// One block (256 threads) per clique. Reps tile (16x256 f32 = 16KB) in LDS.
// ---------------------------------------------------------------------------
__global__ void attn_weights_kernel(const float* __restrict__ reps,
                                    float* __restrict__ w_buf) {
    __shared__ float sx[S_SIZE * D_DIM];   // 16 KB
    __shared__ float sim[S_SIZE][S_SIZE];
    __shared__ float norms[S_SIZE];

    const int c = blockIdx.x;
    const float* R = reps + (size_t)c * S_SIZE * D_DIM;

    for (int idx = threadIdx.x; idx < S_SIZE * D_DIM; idx += blockDim.x)
        sx[idx] = R[idx];
    __syncthreads();

    // 256 threads: thread (i,j) computes dot(reps_i, reps_j) over D=256
    const int i = threadIdx.x >> 4;
    const int j = threadIdx.x & 15;
    float dot = 0.0f;
#pragma unroll 8
    for (int d = 0; d < D_DIM; ++d)
        dot = fmaf(sx[i * D_DIM + d], sx[j * D_DIM + d], dot);
    sim[i][j] = dot;
    __syncthreads();

    if (threadIdx.x < S_SIZE)
        norms[threadIdx.x] = sqrtf(sim[threadIdx.x][threadIdx.x]);
    __syncthreads();

    // cosine similarity with reference's epsilon in the denominator
    float s = sim[i][j] / (norms[i] * norms[j] + EPSF);
    __syncthreads();
    sim[i][j] = s;
    __syncthreads();

    // row softmax (one thread per row; trivially cheap)
    if (threadIdx.x < S_SIZE) {
        const int r = threadIdx.x;
        float m = sim[r][0];
#pragma unroll
        for (int jj = 1; jj < S_SIZE; ++jj) m = fmaxf(m, sim[r][jj]);
        float e[S_SIZE];
        float sum = 0.0f;
#pragma unroll
        for (int jj = 0; jj < S_SIZE; ++jj) {
            e[jj] = expf(sim[r][jj] - m);
            sum += e[jj];
        }
        const float inv = 1.0f / sum;
        float* Wr = w_buf + (size_t)c * S_SIZE * S_SIZE + r * S_SIZE;
#pragma unroll
        for (int jj = 0; jj < S_SIZE; ++jj) Wr[jj] = e[jj] * inv;
    }
}

// ---------------------------------------------------------------------------
// Kernel 2: out[c] = W[c] (16x16) x params[c] (16x65536) via WMMA f32 16x16x4.
// Each wave owns a strip of 16-wide p-tiles for one clique; loop bounds are
// wave-uniform so EXEC stays all-ones for WMMA.
//
// VGPR layouts (ISA 7.12.2, 32-bit operands, wave32):
//   A (16x4):  lane<16 : M=lane,    v0=K(k0+0), v1=K(k0+1)
//              lane>=16: M=lane-16, v0=K(k0+2), v1=K(k0+3)
//   B (4x16):  lane<16 : N=lane,    v0=row(k0+0), v1=row(k0+1)
//              lane>=16: N=lane-16, v0=row(k0+2), v1=row(k0+3)
//   C/D:       VGPR v:  lane<16 -> (M=v,   N=lane); lane>=16 -> (M=v+8, N=lane-16)
// ---------------------------------------------------------------------------
__global__ void wmma_apply_kernel(const float* __restrict__ params,
                                  const float* __restrict__ w_buf,
                                  float* __restrict__ out) {
    const int lane  = threadIdx.x & 31;
    const int half  = lane >> 4;     // 0: lanes 0-15, 1: lanes 16-31
    const int lcol  = lane & 15;

    const int waves_per_block = blockDim.x >> 5;
    const int wave_id   = (int)(blockIdx.x) * waves_per_block + (threadIdx.x >> 5);
    const int n_waves   = (int)(gridDim.x) * waves_per_block;
    const int c         = blockIdx.y;

    // Load A operands (the 16x16 weight matrix, split into four 16x4 slabs).
    const float* Wc = w_buf + (size_t)c * S_SIZE * S_SIZE;   // row-major [i][k]
    v2f a[4];
#pragma unroll
    for (int ka = 0; ka < 4; ++ka) {
        const int k0 = ka * 4 + 2 * half;
        a[ka].x = Wc[lcol * S_SIZE + k0 + 0];
        a[ka].y = Wc[lcol * S_SIZE + k0 + 1];
    }

    const float* Pc = params + (size_t)c * S_SIZE * P_DIM;
    float*       Oc = out    + (size_t)c * S_SIZE * P_DIM;

    const int n_tiles = P_DIM / 16;                 // 4096 tiles per clique
    for (int t = wave_id; t < n_tiles; t += n_waves) {
        const int p0 = t * 16;
        v8f acc = {0.f, 0.f, 0.f, 0.f, 0.f, 0.f, 0.f, 0.f};
#pragma unroll
        for (int ka = 0; ka < 4; ++ka) {
            const int row = ka * 4 + 2 * half;
            v2f b;
            b.x = Pc[(size_t)(row + 0) * P_DIM + p0 + lcol];
            b.y = Pc[(size_t)(row + 1) * P_DIM + p0 + lcol];
            acc = __builtin_amdgcn_wmma_f32_16x16x4_f32(
                false, a[ka], false, b, (short)0, acc, false, false);
        }
#pragma unroll
        for (int v = 0; v < 8; ++v)
            Oc[(size_t)(v + 8 * half) * P_DIM + p0 + lcol] = acc[v];
    }
}

extern "C" void kernel_launch(void* const* d_in, const int* in_sizes, int n_in,
                              void* d_out, int out_size, void* d_ws, size_t ws_size,
                              hipStream_t stream) {
    const float* reps   = (const float*)d_in[0];   // [64,16,256]
    const float* params = (const float*)d_in[1];   // [64,16,65536]
    float*       out    = (float*)d_out;           // [64,16,65536]
    float*       w_buf  = (float*)d_ws;            // 64*16*16 f32 = 64 KB

    attn_weights_kernel<<<dim3(C_CLIQUES), dim3(256), 0, stream>>>(reps, w_buf);

    dim3 grid(64, C_CLIQUES);   // 64 blocks x 8 waves = 512 waves per clique
    wmma_apply_kernel<<<grid, dim3(256), 0, stream>>>(params, w_buf, out);
}